// GATout_17755394802274
// MI455X (gfx1250) — compile-verified
//
#include <hip/hip_runtime.h>

#define D_IN  256
#define D_OUT 16
#define ROWS_PER_BLOCK 32          // 2 waves/block, one 16x16 WMMA tile per wave
#define LDA 260                    // padded LDS row stride (floats): 260 % 64 == 4 -> conflict-free

typedef __attribute__((ext_vector_type(2))) float v2f;
typedef __attribute__((ext_vector_type(4))) float v4f;
typedef __attribute__((ext_vector_type(8))) float v8f;

// ---------------- zero scratch (deg_src | deg_dst | agg are contiguous) ----------------
__global__ void k_zero(float* __restrict__ p, int n) {
    int i = blockIdx.x * blockDim.x + threadIdx.x;
    if (i < n) p[i] = 0.0f;
}

// ---------------- x_lin = x @ W  via V_WMMA_F32_16X16X4_F32 ----------------
// HBM-bound: 205 MB of x streamed once; W (16 KB) stays L2/L0 resident.
__global__ __launch_bounds__(64) void k_gemm(const float* __restrict__ x,
                                             const float* __restrict__ W,
                                             float* __restrict__ xlin,
                                             int n_src) {
    __shared__ float sA[ROWS_PER_BLOCK * LDA];   // 33,280 B
    __shared__ float sW[D_OUT * LDA];            // 16,640 B  (W transposed: sW[n*LDA+k])

    const int  t    = threadIdx.x;          // 0..63
    const int  lane = t & 31;
    const int  wave = t >> 5;               // 0..1
    const long rowBase = (long)blockIdx.x * ROWS_PER_BLOCK;

    // Stage W transposed: W is row-major [256][16]; sW[n*LDA + k] = W[k*16 + n]
    #pragma unroll
    for (int i = 0; i < (D_IN * D_OUT) / (64 * 4); ++i) {   // 16 iters
        int f  = (t + i * 64) * 4;          // flat index, multiple of 4
        int k  = f >> 4;
        int nn = f & 15;
        v4f w4 = *(const v4f*)(W + f);
        sW[(nn + 0) * LDA + k] = w4.x;
        sW[(nn + 1) * LDA + k] = w4.y;
        sW[(nn + 2) * LDA + k] = w4.z;
        sW[(nn + 3) * LDA + k] = w4.w;
    }

    // Stage A tile (32 rows x 256 cols), coalesced float4 global loads
    #pragma unroll
    for (int i = 0; i < (ROWS_PER_BLOCK * D_IN) / (64 * 4); ++i) {  // 32 iters
        int  f4  = t + i * 64;              // float4 index within tile
        int  r   = f4 >> 6;                 // 64 float4 per row
        int  c4  = f4 & 63;
        long row = rowBase + r;
        if (row >= n_src) row = n_src - 1;  // clamp (n_src % 32 == 0 in practice)
        v4f a4 = *(const v4f*)(x + row * D_IN + (c4 << 2));
        *(v4f*)(sA + r * LDA + (c4 << 2)) = a4;   // 16B-aligned: LDA*4 = 1040 = 65*16
    }
    __syncthreads();

    // A frag (16x4 f32): lanes 0-15 hold K=0,1 ; lanes 16-31 hold K=2,3 ; M = lane&15
    // B frag (4x16 f32): N = lane&15 ; lanes 0-15 K=0,1 ; lanes 16-31 K=2,3
    const int   mrow  = (wave << 4) + (lane & 15);
    const int   khalf = (lane >> 4) << 1;            // 0 or 2
    const float* aptr = sA + mrow * LDA + khalf;
    const float* bptr = sW + (lane & 15) * LDA + khalf;

    v8f acc = {};
    #pragma unroll
    for (int kk = 0; kk < D_IN / 4; ++kk) {          // 64 WMMA ops, K=256
        v2f a = *(const v2f*)(aptr + (kk << 2));
        v2f b = *(const v2f*)(bptr + (kk << 2));
        acc = __builtin_amdgcn_wmma_f32_16x16x4_f32(
            /*neg_a=*/false, a, /*neg_b=*/false, b,
            /*c_mod=*/(short)0, acc, /*reuse_a=*/false, /*reuse_b=*/false);
    }

    // C/D layout: lane L, VGPR j -> (M = j + 8*(L>>4), N = L&15)
    const long outRow0 = rowBase + (wave << 4) + ((lane >> 4) << 3);
    const int  nn = lane & 15;
    #pragma unroll
    for (int j = 0; j < 8; ++j) {
        long row = outRow0 + j;
        if (row < n_src) xlin[row * D_OUT + nn] = acc[j];
    }
}

// ---------------- degree counts (L2-resident atomics) ----------------
__global__ void k_degrees(const int* __restrict__ esrc, const int* __restrict__ edst,
                          float* __restrict__ deg_src, float* __restrict__ deg_dst,
                          int n_edges) {
    int e = blockIdx.x * blockDim.x + threadIdx.x;
    if (e < n_edges) {
        atomicAdd(deg_src + esrc[e], 1.0f);
        atomicAdd(deg_dst + edst[e], 1.0f);
    }
}

// ---------------- edge scatter: agg[dst] += norm * x_lin[src]  (4 lanes per edge) ------
__global__ void k_scatter(const int* __restrict__ esrc, const int* __restrict__ edst,
                          const float* __restrict__ xlin,
                          const float* __restrict__ deg_src, const float* __restrict__ deg_dst,
                          float* __restrict__ agg, int n_edges) {
    long tid = (long)blockIdx.x * blockDim.x + threadIdx.x;
    long e = tid >> 2;
    int  c = (int)(tid & 3) << 2;
    if (e >= n_edges) return;
    int s = esrc[e], d = edst[e];
    float norm = rsqrtf((deg_src[s] + 1.0f) * (deg_dst[d] + 1.0f));
    v4f v = *(const v4f*)(xlin + (long)s * D_OUT + c);
    float* dst = agg + (long)d * D_OUT + c;
    atomicAdd(dst + 0, norm * v.x);
    atomicAdd(dst + 1, norm * v.y);
    atomicAdd(dst + 2, norm * v.z);
    atomicAdd(dst + 3, norm * v.w);
}

// ---------------- finalize: + self-loop term + bias, then log_softmax over 16 ----------
__global__ void k_finalize(const float* __restrict__ agg, const float* __restrict__ xlin,
                           const float* __restrict__ deg_dst, const int* __restrict__ res_n_id,
                           const float* __restrict__ bias, float* __restrict__ out,
                           int n_dst) {
    int i = blockIdx.x * blockDim.x + threadIdx.x;
    if (i >= n_dst) return;
    int   nid = res_n_id[i];
    float inv = 1.0f / (deg_dst[i] + 1.0f);
    const float* ap = agg  + (long)i   * D_OUT;
    const float* sp = xlin + (long)nid * D_OUT;
    float v[D_OUT];
    float m = -3.402823466e+38f;
    #pragma unroll
    for (int j = 0; j < D_OUT; ++j) {
        v[j] = ap[j] + sp[j] * inv + bias[j];
        m = fmaxf(m, v[j]);
    }
    float sum = 0.0f;
    #pragma unroll
    for (int j = 0; j < D_OUT; ++j) sum += __expf(v[j] - m);
    float lse = m + __logf(sum);
    float* op = out + (long)i * D_OUT;
    #pragma unroll
    for (int j = 0; j < D_OUT; ++j) op[j] = v[j] - lse;
}

extern "C" void kernel_launch(void* const* d_in, const int* in_sizes, int n_in,
                              void* d_out, int out_size, void* d_ws, size_t ws_size,
                              hipStream_t stream) {
    const float* x    = (const float*)d_in[0];
    const float* W    = (const float*)d_in[1];
    const float* bias = (const float*)d_in[2];
    const int*   esrc = (const int*)d_in[3];
    const int*   edst = (const int*)d_in[4];
    const int*   res  = (const int*)d_in[5];

    const int dout    = in_sizes[2];               // 16
    const int din     = in_sizes[1] / dout;        // 256
    const int n_src   = in_sizes[0] / din;         // 200000
    const int n_edges = in_sizes[3];               // 1600000
    const int n_dst   = in_sizes[5];               // 50000

    float* xlin    = (float*)d_ws;                       // n_src * 16
    float* deg_src = xlin + (size_t)n_src * dout;        // n_src
    float* deg_dst = deg_src + n_src;                    // n_dst
    float* agg     = deg_dst + n_dst;                    // n_dst * 16

    // 1) zero deg_src | deg_dst | agg (contiguous)
    int nz = n_src + n_dst + n_dst * dout;
    k_zero<<<(nz + 255) / 256, 256, 0, stream>>>(deg_src, nz);

    // 2) x_lin = x @ W  (WMMA f32, HBM-bound)
    int gblocks = (n_src + ROWS_PER_BLOCK - 1) / ROWS_PER_BLOCK;
    k_gemm<<<gblocks, 64, 0, stream>>>(x, W, xlin, n_src);

    // 3) degrees
    k_degrees<<<(n_edges + 255) / 256, 256, 0, stream>>>(esrc, edst, deg_src, deg_dst, n_edges);

    // 4) edge scatter (4 lanes/edge)
    long sth = (long)n_edges * 4;
    k_scatter<<<(int)((sth + 255) / 256), 256, 0, stream>>>(esrc, edst, xlin, deg_src, deg_dst, agg, n_edges);

    // 5) finalize + log_softmax
    k_finalize<<<(n_dst + 255) / 256, 256, 0, stream>>>(agg, xlin, deg_dst, res, bias, (float*)d_out, n_dst);
}